// InstancePinOptimizationArea_53558242181775
// MI455X (gfx1250) — compile-verified
//
#include <hip/hip_runtime.h>

// ---------------- Problem constants (from reference) ----------------
constexpr int   NBX         = 1024;
constexpr int   NBY         = 1024;
constexpr int   NUM_NODES   = 4000000;
constexpr int   NUM_PHYS    = 3000000;   // NUM_NODES - NUM_FILLER
constexpr int   NUM_MOVABLE = 2500000;
constexpr float BSX         = 1.0f;      // (XH-XL)/NUM_BINS_X
constexpr float BSY         = 1.0f;
constexpr float STRETCH     = 1.414f;    // PIN_STRETCH_RATIO
constexpr float MAX_ADJ     = 2.5f;
constexpr float MIN_ADJ     = 1.0f / 2.5f;
constexpr float INV_CAP     = 0.5f;      // 1 / UNIT_PIN_CAPACITY  (bsx*bsy == 1)
constexpr int   SPAN        = 4;

static __device__ __forceinline__ int clampi(int v, int lo, int hi) {
    return v < lo ? lo : (v > hi ? hi : v);
}

// ---------------- Kernel 1: zero the 4 MB pin-density map ----------------
__global__ __launch_bounds__(256) void k_zero_pmap(float4* __restrict__ pmap4) {
    int i = blockIdx.x * blockDim.x + threadIdx.x;
    if (i < (NBX * NBY) / 4) {
        pmap4[i] = make_float4(0.0f, 0.0f, 0.0f, 0.0f);
    }
}

// ---------------- Kernel 2: scatter stretched pin boxes into pmap ----------------
// One node per lane. Separable overlaps are computed once per axis (8 instead of
// 16), bounds folded in as zeros; only strictly-positive contributions issue an
// atomic (the reference adds exactly 0 elsewhere). Streams are NT-hinted so the
// L2-resident pmap stays hot.
__global__ __launch_bounds__(256) void k_scatter_pins(
    const float* __restrict__ pos,
    const float* __restrict__ node_size_x,
    const float* __restrict__ node_size_y,
    const int*   __restrict__ pin_weights,
    float*       __restrict__ pmap) {
    int i = blockIdx.x * blockDim.x + threadIdx.x;
    if (i >= NUM_PHYS) return;

    const float sx = __builtin_nontemporal_load(&node_size_x[i]);
    const float sy = __builtin_nontemporal_load(&node_size_y[i]);
    const float px = __builtin_nontemporal_load(&pos[i]);
    const float py = __builtin_nontemporal_load(&pos[NUM_NODES + i]);
    const int   pw = __builtin_nontemporal_load(&pin_weights[i]);

    const float hx = 0.5f * fmaxf(BSX * STRETCH, sx);
    const float hy = 0.5f * fmaxf(BSY * STRETCH, sy);
    const float cx = px + 0.5f * sx;
    const float cy = py + 0.5f * sy;
    const float xmin = cx - hx, xmax = cx + hx;
    const float ymin = cy - hy, ymax = cy + hy;
    const float dens = (float)pw / (4.0f * hx * hy);

    const int bxl = clampi((int)floorf(xmin), 0, NBX - 1);
    const int byl = clampi((int)floorf(ymin), 0, NBY - 1);

    // Separable overlap vectors with bounds folded in (0 => skip).
    float dox[SPAN], oyv[SPAN];
#pragma unroll
    for (int k = 0; k < SPAN; ++k) {
        const int bx = bxl + k;
        const float ox = fminf(xmax, (float)(bx + 1)) - fmaxf(xmin, (float)bx);
        dox[k] = (bx < NBX && ox > 0.0f) ? dens * ox : 0.0f;  // (dens*ox) first: matches ref assoc.
        const int by = byl + k;
        const float oy = fminf(ymax, (float)(by + 1)) - fmaxf(ymin, (float)by);
        oyv[k] = (by < NBY && oy > 0.0f) ? oy : 0.0f;
    }

#pragma unroll
    for (int di = 0; di < SPAN; ++di) {
        if (dox[di] <= 0.0f) continue;
        float* __restrict__ row = pmap + (bxl + di) * NBY;
#pragma unroll
        for (int dj = 0; dj < SPAN; ++dj) {
            const float val = dox[di] * oyv[dj];          // ((dens*ox)*oy) as in reference
            if (val > 0.0f) {
                atomicAdd(&row[byl + dj], val);           // global_atomic_add_f32, no return
            }
        }
    }
}

// ---------------- Kernel 3: gather clipped adjustment, fused normalize/clip ----------------
__global__ __launch_bounds__(256) void k_gather_area(
    const float* __restrict__ pos,
    const float* __restrict__ node_size_x,
    const float* __restrict__ node_size_y,
    const float* __restrict__ pmap,
    float*       __restrict__ area) {
    int i = blockIdx.x * blockDim.x + threadIdx.x;
    if (i >= NUM_MOVABLE) return;

    const float mx  = __builtin_nontemporal_load(&pos[i]);
    const float my  = __builtin_nontemporal_load(&pos[NUM_NODES + i]);
    const float msx = __builtin_nontemporal_load(&node_size_x[i]);
    const float msy = __builtin_nontemporal_load(&node_size_y[i]);

    const float xmax = mx + msx;
    const float ymax = my + msy;
    const int bxl = clampi((int)floorf(mx), 0, NBX - 1);
    const int byl = clampi((int)floorf(my), 0, NBY - 1);

    // Separable overlap vectors with bounds folded in (0 => contributes nothing).
    float oxv[SPAN], oyv[SPAN];
#pragma unroll
    for (int k = 0; k < SPAN; ++k) {
        const int bx = bxl + k;
        const float ox = fminf(xmax, (float)(bx + 1)) - fmaxf(mx, (float)bx);
        oxv[k] = (bx < NBX && ox > 0.0f) ? ox : 0.0f;
        const int by = byl + k;
        const float oy = fminf(ymax, (float)(by + 1)) - fmaxf(my, (float)by);
        oyv[k] = (by < NBY && oy > 0.0f) ? oy : 0.0f;
    }

    float acc = 0.0f;
#pragma unroll
    for (int di = 0; di < SPAN; ++di) {
        if (oxv[di] <= 0.0f) continue;
        const float* __restrict__ row = pmap + (bxl + di) * NBY;
#pragma unroll
        for (int dj = 0; dj < SPAN; ++dj) {
            const float w = oxv[di] * oyv[dj];            // (ox*oy) first: matches ref assoc.
            if (w > 0.0f) {
                // adj = clip(pmap / UNIT_PIN_CAPACITY, MIN_ADJ, MAX_ADJ); bsx*bsy == 1
                const float adj = fminf(fmaxf(row[byl + dj] * INV_CAP, MIN_ADJ), MAX_ADJ);
                acc += w * adj;                           // separate mul+add as in reference
            }
        }
    }
    __builtin_nontemporal_store(acc, &area[i]);
}

// ---------------- Host-side launcher ----------------
extern "C" void kernel_launch(void* const* d_in, const int* in_sizes, int n_in,
                              void* d_out, int out_size, void* d_ws, size_t ws_size,
                              hipStream_t stream) {
    const float* pos          = (const float*)d_in[0];  // 2*NUM_NODES
    const float* node_size_x  = (const float*)d_in[1];  // NUM_NODES
    const float* node_size_y  = (const float*)d_in[2];  // NUM_NODES
    const int*   pin_weights  = (const int*)  d_in[3];  // NUM_PHYS
    float*       out          = (float*)d_out;          // NUM_MOVABLE
    float*       pmap         = (float*)d_ws;           // NBX*NBY floats (4 MB)

    const dim3 blk(256);

    // Zero the density map every call (workspace is poisoned by the harness).
    k_zero_pmap<<<dim3(((NBX * NBY) / 4 + 255) / 256), blk, 0, stream>>>((float4*)pmap);

    // Phase 1: scatter pin density (L2-resident atomics).
    k_scatter_pins<<<dim3((NUM_PHYS + 255) / 256), blk, 0, stream>>>(
        pos, node_size_x, node_size_y, pin_weights, pmap);

    // Phase 2: fused normalize/clip + overlap-weighted gather.
    k_gather_area<<<dim3((NUM_MOVABLE + 255) / 256), blk, 0, stream>>>(
        pos, node_size_x, node_size_y, pmap, out);
}